// Attention_6047313953472
// MI455X (gfx1250) — compile-verified
//
#include <hip/hip_runtime.h>

typedef __attribute__((ext_vector_type(2))) float v2f;
typedef __attribute__((ext_vector_type(8))) float v8f;

#define DIM 64

// Monotone mapping float -> uint so unsigned atomicMax == float max.
__device__ __forceinline__ unsigned f2ord(float f) {
    unsigned u = __float_as_uint(f);
    return (u & 0x80000000u) ? ~u : (u | 0x80000000u);
}
__device__ __forceinline__ float ord2f(unsigned u) {
    return (u & 0x80000000u) ? __uint_as_float(u & 0x7FFFFFFFu)
                             : __uint_as_float(~u);
}

// ws layout: [0, NE)           : e scores (float)
//            [NE, NE+NS)       : mapped segment max (uint32)
//            [NE+NS, NE+2*NS)  : segment sums (float)
// NS = num_nodes * heads is only known on-device (num_nodes is a device scalar),
// so every kernel recomputes the pointers from the num_nodes pointer.

__global__ void init_seg(float* __restrict__ ws, const int* __restrict__ num_nodes_ptr,
                         int heads, int NE) {
    long long num_seg = (long long)(*num_nodes_ptr) * heads;
    unsigned* m = (unsigned*)(ws + NE);
    float* s = (float*)(m + num_seg);
    long long stride = (long long)gridDim.x * blockDim.x;
    for (long long i = (long long)blockIdx.x * blockDim.x + threadIdx.x;
         i < num_seg; i += stride) {
        m[i] = 0u;      // maps below every finite float
        s[i] = 0.0f;
    }
}

// One wave = 16 edges of one head. C = A(16x4) x B(4x16) accumulated over 16
// K-steps (DIM=64) with V_WMMA_F32_16X16X4_F32.
// A layout: lane L<16 holds (M=L, K=0..1), lane L+16 holds (M=L, K=2..3).
// B: column 0 = a_l chunk, column 1 = a_r chunk, other columns zero.
//    All 32 lanes load unconditionally from a clamped address (lanes n>=2
//    reuse the n==1 address, fully cached) and zero via a per-lane multiplier
//    -- no EXEC branching inside the WMMA loop.
// C layout: element (M,N) lives in VGPR M%8, lane N + 16*(M/8).
__global__ void __launch_bounds__(256)
gat_scores(const float* __restrict__ x_i, const float* __restrict__ x_j,
           const float* __restrict__ a, const int* __restrict__ edge_index,
           float* __restrict__ ws, int E, int NE) {
    __shared__ float sc[8][2][32][8];   // [wave][matrix i/j][lane][vgpr] = 16 KB

    const int lane = threadIdx.x & 31;
    const int wave = threadIdx.x >> 5;
    long long tile = (long long)blockIdx.x * 8 + wave;
    long long base = tile * 16;
    const bool active = (base < NE);
    long long cbase = active ? base : 0;         // clamp so loads stay in-bounds

    const int h    = (int)(cbase / E);           // E % 16 == 0: tile in one head
    const int n    = lane & 15;                  // M (A) / N (B)
    const int half = lane >> 4;                  // selects K pair {0,1} vs {2,3}

    const float* xi_p = x_i + (cbase + n) * (long long)DIM + 2 * half;
    const float* xj_p = x_j + (cbase + n) * (long long)DIM + 2 * half;
    // n==0 -> a_l, n>=1 -> a_r (clamped); zeroed below for n>=2.
    const float* a_p = a + (long long)h * 2 * DIM + (n < 1 ? 0 : DIM) + 2 * half;
    const float bmask = (n < 2) ? 1.0f : 0.0f;

    v8f ci = {};
    v8f cj = {};
#pragma unroll
    for (int t = 0; t < 16; ++t) {
        const int k = 4 * t;
        v2f av_i = *(const v2f*)(xi_p + k);
        v2f av_j = *(const v2f*)(xj_p + k);
        v2f bv   = *(const v2f*)(a_p + k);
        bv.x *= bmask;
        bv.y *= bmask;
        ci = __builtin_amdgcn_wmma_f32_16x16x4_f32(false, av_i, false, bv,
                                                   (short)0, ci, false, false);
        cj = __builtin_amdgcn_wmma_f32_16x16x4_f32(false, av_j, false, bv,
                                                   (short)0, cj, false, false);
    }

#pragma unroll
    for (int r = 0; r < 8; ++r) {
        sc[wave][0][lane][r] = ci[r];
        sc[wave][1][lane][r] = cj[r];
    }
    __syncthreads();

    if (active && lane < 16) {
        const int grp = lane >> 3;       // M/8 -> lane-half of accumulator
        const int idx = lane & 7;        // M%8 -> VGPR component
        float ci_l = sc[wave][0][grp * 16 + 0][idx];   // xi . a_l
        float ci_r = sc[wave][0][grp * 16 + 1][idx];   // xi . a_r
        float cj_l = sc[wave][1][grp * 16 + 0][idx];   // xj . a_l
        float cj_r = sc[wave][1][grp * 16 + 1][idx];   // xj . a_r
        float e_l = ci_l + cj_r;
        float e_r = cj_l + ci_r;
        float el = (e_l > 0.0f) ? e_l : 0.2f * e_l;    // leaky_relu 0.2
        float er = (e_r > 0.0f) ? e_r : 0.2f * e_r;
        float e = el + er;

        long long g = base + lane;
        ws[g] = e;
        int seg = edge_index[(long long)NE + g];       // edge_index[1][g]
        unsigned* m = (unsigned*)(ws + NE);
        atomicMax(&m[seg], f2ord(e));
    }
}

__global__ void seg_expsum(const int* __restrict__ edge_index,
                           const int* __restrict__ num_nodes_ptr,
                           float* __restrict__ ws, int heads, int NE) {
    long long num_seg = (long long)(*num_nodes_ptr) * heads;
    unsigned* m = (unsigned*)(ws + NE);
    float* s = (float*)(m + num_seg);
    long long g = (long long)blockIdx.x * blockDim.x + threadIdx.x;
    if (g >= NE) return;
    int seg = edge_index[(long long)NE + g];
    float mx = ord2f(m[seg]);
    float ex = __expf(ws[g] - mx);
    atomicAdd(&s[seg], ex);
}

__global__ void seg_norm(const int* __restrict__ edge_index,
                         const int* __restrict__ num_nodes_ptr,
                         const float* __restrict__ ws, float* __restrict__ out,
                         int heads, int NE) {
    long long num_seg = (long long)(*num_nodes_ptr) * heads;
    const unsigned* m = (const unsigned*)(ws + NE);
    const float* s = (const float*)(m + num_seg);
    long long g = (long long)blockIdx.x * blockDim.x + threadIdx.x;
    if (g >= NE) return;
    int seg = edge_index[(long long)NE + g];
    float mx = ord2f(m[seg]);
    float ex = __expf(ws[g] - mx);
    out[g] = ex / (s[seg] + 1e-16f);
}

extern "C" void kernel_launch(void* const* d_in, const int* in_sizes, int n_in,
                              void* d_out, int out_size, void* d_ws, size_t ws_size,
                              hipStream_t stream) {
    const float* x_i       = (const float*)d_in[0];
    const float* x_j       = (const float*)d_in[1];
    const float* a         = (const float*)d_in[2];
    const int*   edge_idx  = (const int*)d_in[3];
    const int*   num_nodes = (const int*)d_in[4];

    const int NE    = in_sizes[0] / DIM;        // heads * E = 1.6M
    const int heads = in_sizes[2] / (2 * DIM);  // 8
    const int E     = NE / heads;               // 200000

    float* ws  = (float*)d_ws;
    float* out = (float*)d_out;

    init_seg<<<2048, 256, 0, stream>>>(ws, num_nodes, heads, NE);

    const int tiles  = (NE + 15) / 16;
    const int blocks = (tiles + 7) / 8;          // 8 waves (tiles) per block
    gat_scores<<<blocks, 256, 0, stream>>>(x_i, x_j, a, edge_idx, ws, E, NE);

    const int b2 = (NE + 255) / 256;
    seg_expsum<<<b2, 256, 0, stream>>>(edge_idx, num_nodes, ws, heads, NE);
    seg_norm<<<b2, 256, 0, stream>>>(edge_idx, num_nodes, ws, out, heads, NE);
}